// LearnedHMFEncoder_58291296141932
// MI455X (gfx1250) — compile-verified
//
#include <hip/hip_runtime.h>
#include <cstdint>
#include <cstddef>

// Problem constants (reference): B=8192, N=256, D=6, R=3, H=32
#define BATCHES 8192
#define NNBR    256
#define DDIM    6
#define RROLE   3
#define HHID    32
#define OUTD    (RROLE * DDIM)   // 18

typedef __attribute__((ext_vector_type(2))) float v2f;
typedef __attribute__((ext_vector_type(8))) float v8f;

__device__ __forceinline__ float wave_sum(float x) {
#pragma unroll
  for (int o = 16; o > 0; o >>= 1) x += __shfl_xor(x, o, 32);
  return x;
}
__device__ __forceinline__ float wave_max(float x) {
#pragma unroll
  for (int o = 16; o > 0; o >>= 1) x = fmaxf(x, __shfl_xor(x, o, 32));
  return x;
}

// Single-instruction relu: v_med3(x, 0, +inf) — avoids the canonicalize
// v_max_num pair that fmaxf(x, 0.f) lowers to.
__device__ __forceinline__ float relu1(float x) {
  return __builtin_amdgcn_fmed3f(x, 0.0f, __builtin_inff());
}

// One wave32 per batch. 8 waves (256 threads) per block, 1024 blocks.
__global__ __launch_bounds__(256)
void hmf_encoder_kernel(const float* __restrict__ S,        // [B,N,D]
                        const int* __restrict__ roles,      // [B,N]
                        const unsigned char* __restrict__ maskp, // [B,N] bool (1B)
                        const float* __restrict__ trust,    // [B,N]
                        const float* __restrict__ W1,       // [R,D,H]
                        const float* __restrict__ b1,       // [R,H]
                        const float* __restrict__ W2,       // [R,H]
                        const float* __restrict__ b2,       // [R]
                        float* __restrict__ out) {          // [B, R*D]
  __shared__ float lds_ms[8 * RROLE * NNBR];  // masked scores (0 where inactive)

  const int lane = threadIdx.x & 31;
  const int wv   = threadIdx.x >> 5;
  const int b    = blockIdx.x * 8 + wv;
  const int L    = lane & 15;
  const bool hi  = lane >= 16;

  const float*         Sb     = S      + (size_t)b * (NNBR * DDIM);
  const int*           rolesb = roles  + (size_t)b * NNBR;
  const unsigned char* maskb  = maskp  + (size_t)b * NNBR;
  const float*         trustb = trust  + (size_t)b * NNBR;

  // Pull this batch's 6KB state slab toward the caches (global_prefetch_b8).
  __builtin_prefetch(Sb + lane * 48, 0, 1);

  // ---------------- per-wave constant setup ----------------
  // A = W1cat^T : [96 x 8], row m = r*32+h, col k: k<6 -> W1[r][k][h],
  // k==6 -> b1[r][h] (bias row), k==7 -> 0 (pad).
  // ISA 16x4 f32 A layout: lanes 0-15 hold {K=4s, 4s+1}, lanes 16-31 {4s+2, 4s+3}.
  v2f a[6][2];
  float w2s[6][8];
#pragma unroll
  for (int i = 0; i < 6; ++i) {
    int m = 16 * i + L;
    int r = m >> 5;
    int h = m & 31;
    const float* w1r = W1 + (size_t)(r * 6) * 32 + h;  // element k at w1r[k*32]
    if (!hi) {
      a[i][0] = (v2f){w1r[0 * 32], w1r[1 * 32]};
      a[i][1] = (v2f){w1r[4 * 32], w1r[5 * 32]};
    } else {
      a[i][0] = (v2f){w1r[2 * 32], w1r[3 * 32]};
      a[i][1] = (v2f){b1[r * 32 + h], 0.0f};           // bias row K=6, pad K=7
    }
    // W2 coefficient for the h-value living in VGPR v of tile i on this lane:
    // C/D layout: lanes 0-15 -> M=v, lanes 16-31 -> M=v+8; flat W2 index == m.
#pragma unroll
    for (int v = 0; v < 8; ++v)
      w2s[i][v] = W2[16 * i + v + (hi ? 8 : 0)];
  }
  const float b2v0 = b2[0], b2v1 = b2[1], b2v2 = b2[2];

  float* msw = lds_ms + wv * (RROLE * NNBR);

  // ---------------- pass 1: all-role scores via f32 WMMA ----------------
  for (int t = 0; t < 16; ++t) {
    int n = t * 16 + L;
    const float* row = Sb + n * DDIM;
    // B = S^T tile [8 x 16]: lane owns column n. k-step0: {d0,d1}/{d2,d3};
    // k-step1: {d4,d5} on low lanes, bias row {1,0} on high lanes.
    v2f bk0 = *(const v2f*)(row + (hi ? 2 : 0));
    v2f s45 = *(const v2f*)(row + 4);       // unconditional load (always in-bounds)
    v2f bk1;                                // branchless select -> v_cndmask, no EXEC branch
    bk1.x = hi ? 1.0f : s45.x;
    bk1.y = hi ? 0.0f : s45.y;

    float p0 = 0.f, p1 = 0.f, p2 = 0.f;
#pragma unroll
    for (int i = 0; i < 6; ++i) {
      v8f c = {};
      c = __builtin_amdgcn_wmma_f32_16x16x4_f32(false, a[i][0], false, bk0,
                                                (short)0, c, false, false);
      c = __builtin_amdgcn_wmma_f32_16x16x4_f32(false, a[i][1], false, bk1,
                                                (short)0, c, false, false);
      float pr = 0.f;
#pragma unroll
      for (int v = 0; v < 8; ++v)
        pr = fmaf(relu1(c[v]), w2s[i][v], pr);         // relu + h·W2 partial
      if (i < 2)      p0 += pr;
      else if (i < 4) p1 += pr;
      else            p2 += pr;
    }
    // Each neighbor's score is split across lane L (M%16<8) and L+16 (M%16>=8).
    p0 += __shfl_xor(p0, 16, 32);
    p1 += __shfl_xor(p1, 16, 32);
    p2 += __shfl_xor(p2, 16, 32);

    if (!hi) {
      int   role = rolesb[n];
      float mk   = maskb[n] ? 1.0f : 0.0f;
      float sc[3] = {p0 + b2v0, p1 + b2v1, p2 + b2v2};
#pragma unroll
      for (int r = 0; r < 3; ++r) {
        float act = (role == r) ? mk : 0.0f;
        msw[r * NNBR + n] = act * sc[r];  // multiplicative masking, as reference
      }
    }
  }

  // LDS RAW fence (this wave only) — CDNA5 split dependency counter.
  asm volatile("s_wait_dscnt 0" ::: "memory");

  // ---------------- pass 2: masked softmax + trust-weighted mean ----------------
  float mx[3];
#pragma unroll
  for (int r = 0; r < 3; ++r) {
    float m = -3.4e38f;
#pragma unroll
    for (int j = 0; j < 8; ++j)
      m = fmaxf(m, msw[r * NNBR + 32 * j + lane]);  // includes the zeros, as ref
    mx[r] = wave_max(m);
  }

  float eS[3] = {0.f, 0.f, 0.f};   // sum of e        -> denom
  float T [3] = {0.f, 0.f, 0.f};   // sum of e*trust  -> ws*denom
  float U [3][6] = {};             // sum of e*trust*S[d]
  for (int j = 0; j < 8; ++j) {
    int n = 32 * j + lane;
    float tr   = trustb[n];
    int   role = rolesb[n];                        // L0-resident reload,
    float mk   = maskb[n] ? 1.0f : 0.0f;           // replaces lds_rm array
    const float* row = Sb + n * DDIM;
    v2f s01 = *(const v2f*)(row);
    v2f s23 = *(const v2f*)(row + 2);
    v2f s45 = *(const v2f*)(row + 4);
#pragma unroll
    for (int r = 0; r < 3; ++r) {
      float act = (role == r) ? mk : 0.0f;
      float e = act * __expf(msw[r * NNBR + n] - mx[r]);
      eS[r] += e;
      float et = e * tr;
      T[r] += et;
      U[r][0] = fmaf(et, s01.x, U[r][0]);
      U[r][1] = fmaf(et, s01.y, U[r][1]);
      U[r][2] = fmaf(et, s23.x, U[r][2]);
      U[r][3] = fmaf(et, s23.y, U[r][3]);
      U[r][4] = fmaf(et, s45.x, U[r][4]);
      U[r][5] = fmaf(et, s45.y, U[r][5]);
    }
  }
#pragma unroll
  for (int r = 0; r < 3; ++r) {
    eS[r] = wave_sum(eS[r]);
    T[r]  = wave_sum(T[r]);
#pragma unroll
    for (int d = 0; d < 6; ++d) U[r][d] = wave_sum(U[r][d]);
  }

  if (lane == 0) {
    // 18 consecutive floats from one lane -> compiler merges into b128 stores.
#pragma unroll
    for (int r = 0; r < 3; ++r) {
      float denom = eS[r] + 1e-8f;
      float ws    = fmaxf(T[r] / denom, 1e-8f);      // clip(sum w, 1e-8)
      float inv   = 1.0f / (denom * ws);
#pragma unroll
      for (int d = 0; d < 6; ++d)
        out[(size_t)b * OUTD + r * 6 + d] = U[r][d] * inv;
    }
  }
}

extern "C" void kernel_launch(void* const* d_in, const int* in_sizes, int n_in,
                              void* d_out, int out_size, void* d_ws, size_t ws_size,
                              hipStream_t stream) {
  (void)in_sizes; (void)n_in; (void)out_size; (void)d_ws; (void)ws_size;
  const float*         S      = (const float*)d_in[0];
  const int*           roles  = (const int*)d_in[1];
  const unsigned char* maskp  = (const unsigned char*)d_in[2];  // jnp.bool_ = 1 byte
  const float*         trust  = (const float*)d_in[3];
  const float*         W1     = (const float*)d_in[4];
  const float*         b1     = (const float*)d_in[5];
  const float*         W2     = (const float*)d_in[6];
  const float*         b2     = (const float*)d_in[7];
  float*               out    = (float*)d_out;

  dim3 grid(BATCHES / 8);   // 1024 blocks, 8 waves (batches) each
  dim3 block(256);
  hipLaunchKernelGGL(hmf_encoder_kernel, grid, block, 0, stream,
                     S, roles, maskp, trust, W1, b1, W2, b2, out);
}